// JTNNDecoder_69123203662616
// MI455X (gfx1250) — compile-verified
//
#include <hip/hip_runtime.h>
#include <math.h>

#define KT 32
#define KN 1024
#define KH 256
#define KL 128
#define KV 800
#define KB 256
#define KNB 15
#define KE (KT*KN)    /* 32768 */
#define KEB (KE+KB)   /* 33024 */

typedef __attribute__((ext_vector_type(2))) float v2f;
typedef __attribute__((ext_vector_type(8))) float v8f;

enum { M_NONE = 0, M_RELU = 1, M_SIG = 2, M_RGATE = 3, M_GRU = 4 };

// ---------------------------------------------------------------------------
// fp32 WMMA GEMM:  Y[row+yRowOff, col] = epilogue( X @ W^T + bias )
// X: [rows, K] (leading dim ldx), W: [outDim, K] row-major.
// One wave computes a 64x32 macro-tile = 4x2 grid of 16x16 WMMA tiles
// (8 f32 accumulators, 64 acc VGPRs/lane). Per K=4 chunk: 4 A-loads +
// 2 B-loads feed 8 v_wmma_f32_16x16x4_f32 -> 5.3 FLOP per loaded byte
// and 8 independent accumulation chains for XDL latency hiding.
// A tile layout (ISA 7.12.2): lane L holds {A[L&15][kb], A[L&15][kb+1]},
// kb = (L>>4)*2 ; B mirrored ; C/D: VGPR v -> row v + 8*(L>>4), col L&15.
// ---------------------------------------------------------------------------
__global__ void __launch_bounds__(32)
gemm64x32(const float* __restrict__ X, int ldx,
          const float* __restrict__ W,
          const float* __restrict__ bias,
          float* __restrict__ Y, int ldy, int yRowOff,
          int K, int mode,
          const float* __restrict__ aux0,
          const float* __restrict__ aux1)
{
    const int lane  = threadIdx.x;
    const int m     = lane & 15;
    const int half  = lane >> 4;
    const int kb    = half << 1;
    const int rbase = blockIdx.x * 64;
    const int cbase = blockIdx.y * 32;

    const float* xr[4];
#pragma unroll
    for (int rt = 0; rt < 4; ++rt)
        xr[rt] = X + (long)(rbase + rt * 16 + m) * ldx + kb;
    const float* wr[2];
#pragma unroll
    for (int ct = 0; ct < 2; ++ct)
        wr[ct] = W + (long)(cbase + ct * 16 + m) * K + kb;

    const v8f vzero = {0.f, 0.f, 0.f, 0.f, 0.f, 0.f, 0.f, 0.f};
    v8f acc[4][2];
#pragma unroll
    for (int rt = 0; rt < 4; ++rt)
#pragma unroll
        for (int ct = 0; ct < 2; ++ct)
            acc[rt][ct] = vzero;

#pragma unroll 2
    for (int k = 0; k < K; k += 4) {
        v2f b0, b1;
        b0.x = wr[0][k]; b0.y = wr[0][k + 1];
        b1.x = wr[1][k]; b1.y = wr[1][k + 1];
#pragma unroll
        for (int rt = 0; rt < 4; ++rt) {
            v2f a;
            a.x = xr[rt][k]; a.y = xr[rt][k + 1];
            acc[rt][0] = __builtin_amdgcn_wmma_f32_16x16x4_f32(
                false, a, false, b0, (short)0, acc[rt][0], false, false);
            acc[rt][1] = __builtin_amdgcn_wmma_f32_16x16x4_f32(
                false, a, false, b1, (short)0, acc[rt][1], false, false);
        }
    }

#pragma unroll
    for (int rt = 0; rt < 4; ++rt) {
#pragma unroll
        for (int ct = 0; ct < 2; ++ct) {
            const int col = cbase + ct * 16 + m;
            const float bv = bias ? bias[col] : 0.0f;
#pragma unroll
            for (int v = 0; v < 8; ++v) {
                const int row = rbase + rt * 16 + v + (half << 3);
                float val = acc[rt][ct][v] + bv;
                float* yp = Y + (long)(row + yRowOff) * ldy + col;
                switch (mode) {
                case M_NONE: *yp = val; break;
                case M_RELU: *yp = fmaxf(val, 0.f); break;
                case M_SIG:  *yp = 1.f / (1.f + expf(-val)); break;
                case M_RGATE: {
                    // val = (h_nei @ U_r^T); aux0 = rx (x@W_r^T+b), aux1 = h_nei
                    float r = 1.f / (1.f + expf(-(val + aux0[(long)(row / KNB) * KH + col])));
                    *yp = r * aux1[(long)row * KH + col];
                } break;
                case M_GRU: {
                    // val = pre-activation h_tilde; aux0 = z, aux1 = sum_h
                    float ht = tanhf(val);
                    float z  = aux0[(long)row * KH + col];
                    float sh = aux1[(long)row * KH + col];
                    *yp = (1.f - z) * sh + z * ht;
                } break;
                }
            }
        }
    }
}

// ---------------------------------------------------------------------------
// Per-step gathers: x = emb[wid], h_nei gather + sum, cur_o sum, write
// catA = [x | sum_h], sumh, hnei, and the stop-head row [x | cur_o].
// ---------------------------------------------------------------------------
__global__ void __launch_bounds__(256)
gru_gather(const float* __restrict__ emb, const float* __restrict__ hbuf,
           const int* __restrict__ wid_t, const int* __restrict__ hidx_t,
           const int* __restrict__ oidx_t,
           float* __restrict__ catA, float* __restrict__ sumh,
           float* __restrict__ hnei, float* __restrict__ stoprow)
{
    const int n = blockIdx.x, c = threadIdx.x;
    const float x = emb[(long)wid_t[n] * KH + c];
    float sh = 0.f, co = 0.f;
#pragma unroll
    for (int j = 0; j < KNB; ++j) {
        const int hi = hidx_t[n * KNB + j];
        const float v = hbuf[(long)hi * KH + c];
        hnei[(long)(n * KNB + j) * KH + c] = v;
        sh += v;
        const int oi = oidx_t[n * KNB + j];
        co += hbuf[(long)oi * KH + c];
    }
    catA[(long)n * (2 * KH) + c] = x;
    catA[(long)n * (2 * KH) + KH + c] = sh;
    sumh[(long)n * KH + c] = sh;
    stoprow[(long)n * (2 * KH) + c] = x;
    stoprow[(long)n * (2 * KH) + KH + c] = co;
}

// gated[n] = sum_j (r*h_nei)[n,j]; written into catA second half (W_h input)
__global__ void __launch_bounds__(256)
gated_reduce(const float* __restrict__ rh, float* __restrict__ catA)
{
    const int n = blockIdx.x, c = threadIdx.x;
    float g = 0.f;
#pragma unroll
    for (int j = 0; j < KNB; ++j) g += rh[(long)(n * KNB + j) * KH + c];
    catA[(long)n * (2 * KH) + KH + c] = g;
}

// predcat[i] = [ hid_i | x_tree[ctx_i] ]  (rows 0..B-1: zero hid, ctx=i)
__global__ void __launch_bounds__(384)
pred_gather(const float* __restrict__ hbuf, const float* __restrict__ xtree,
            const int* __restrict__ ctxs, float* __restrict__ predcat)
{
    const long i = blockIdx.x; const int c = threadIdx.x;
    float v;
    if (i < KB) {
        v = (c < KH) ? 0.f : xtree[i * KL + (c - KH)];
    } else {
        const long e = i - KB;
        v = (c < KH) ? hbuf[e * KH + c]
                     : xtree[(long)ctxs[e] * KL + (c - KH)];
    }
    predcat[i * (KH + KL) + c] = v;
}

// stopcat[i] = [ s1_i | x_tree[ctx_i] ]  (rows E..E+B-1: ctx = i-E)
__global__ void __launch_bounds__(384)
stop_gather(const float* __restrict__ s1, const float* __restrict__ xtree,
            const int* __restrict__ ctxs, float* __restrict__ stopcat)
{
    const long i = blockIdx.x; const int c = threadIdx.x;
    const int ctx = (i < KE) ? ctxs[i] : (int)(i - KE);
    const float v = (c < KH) ? s1[i * KH + c]
                             : xtree[(long)ctx * KL + (c - KH)];
    stopcat[i * (KH + KL) + c] = v;
}

// root closing rows of the stop buffer: [emb[root_wid] | sum_j hbuf[root_o_idx]]
__global__ void __launch_bounds__(256)
root_gather(const float* __restrict__ emb, const float* __restrict__ hbuf,
            const int* __restrict__ rwid, const int* __restrict__ roidx,
            float* __restrict__ stopbuf)
{
    const int b = blockIdx.x, c = threadIdx.x;
    const float x = emb[(long)rwid[b] * KH + c];
    float co = 0.f;
#pragma unroll
    for (int j = 0; j < KNB; ++j)
        co += hbuf[(long)roidx[b * KNB + j] * KH + c];
    const long row = (long)KE + b;
    stopbuf[row * (2 * KH) + c] = x;
    stopbuf[row * (2 * KH) + KH + c] = co;
}

// Per-row log-softmax NLL + argmax accuracy (V=800), masked accumulation.
__global__ void __launch_bounds__(256)
pred_loss_kernel(const float* __restrict__ scores,
                 const int* __restrict__ pred_targets,
                 const int* __restrict__ direction,
                 const int* __restrict__ root_word_ids,
                 float* __restrict__ accum)
{
    __shared__ float smax[256];
    __shared__ int   sarg[256];
    __shared__ float ssum[256];
    const int tid = threadIdx.x;
    const long i = blockIdx.x;
    const float* s = scores + i * KV;

    float mx = -3.4e38f; int am = KV;
    for (int j = tid; j < KV; j += 256) {
        const float v = s[j];
        if (v > mx || (v == mx && j < am)) { mx = v; am = j; }
    }
    smax[tid] = mx; sarg[tid] = am; __syncthreads();
    for (int st = 128; st > 0; st >>= 1) {
        if (tid < st) {
            const float v2 = smax[tid + st]; const int a2 = sarg[tid + st];
            if (v2 > smax[tid] || (v2 == smax[tid] && a2 < sarg[tid])) {
                smax[tid] = v2; sarg[tid] = a2;
            }
        }
        __syncthreads();
    }
    const float Mx = smax[0]; const int Am = sarg[0];

    float p = 0.f;
    for (int j = tid; j < KV; j += 256) p += expf(s[j] - Mx);
    ssum[tid] = p; __syncthreads();
    for (int st = 128; st > 0; st >>= 1) {
        if (tid < st) ssum[tid] += ssum[tid + st];
        __syncthreads();
    }

    if (tid == 0) {
        int tgt; float mask;
        if (i < KB) { tgt = root_word_ids[i]; mask = 1.f; }
        else        { tgt = pred_targets[i - KB]; mask = (float)direction[i - KB]; }
        const float lse = Mx + logf(ssum[0]);
        const float nll = lse - s[tgt];
        atomicAdd(&accum[0], nll * mask);
        if (Am == tgt) atomicAdd(&accum[1], mask);
        atomicAdd(&accum[2], mask);
    }
}

// s = s2[i] . Uo_w + Uo_b ; BCE-with-logits + accuracy accumulation.
__global__ void __launch_bounds__(256)
stop_loss_kernel(const float* __restrict__ s2, const float* __restrict__ Uo_w,
                 const float* __restrict__ Uo_b,
                 const int* __restrict__ direction, float* __restrict__ accum)
{
    __shared__ float red[256];
    const int tid = threadIdx.x;
    const long i = blockIdx.x;
    red[tid] = s2[i * KH + tid] * Uo_w[tid];
    __syncthreads();
    for (int st = 128; st > 0; st >>= 1) {
        if (tid < st) red[tid] += red[tid + st];
        __syncthreads();
    }
    if (tid == 0) {
        const float s = red[0] + Uo_b[0];
        const float tgt = (i < KE) ? (float)direction[i] : 0.f;
        const float bce = fmaxf(s, 0.f) - s * tgt + log1pf(expf(-fabsf(s)));
        atomicAdd(&accum[3], bce);
        const float pr = (s >= 0.f) ? 1.f : 0.f;
        if (pr == tgt) atomicAdd(&accum[4], 1.f);
    }
}

__global__ void zero_init(float* __restrict__ hbuf, float* __restrict__ accum)
{
    const int c = threadIdx.x;
    hbuf[(long)KE * KH + c] = 0.f;  // zero-pad message row
    if (c < 8) accum[c] = 0.f;
}

__global__ void finalize_kernel(const float* __restrict__ accum, float* __restrict__ out)
{
    out[0] = accum[0] / (float)KB;   // pred_loss
    out[1] = accum[3] / (float)KB;   // stop_loss
    out[2] = accum[1] / accum[2];    // pred_acc
    out[3] = accum[4] / (float)KEB;  // stop_acc
}

extern "C" void kernel_launch(void* const* d_in, const int* in_sizes, int n_in,
                              void* d_out, int out_size, void* d_ws, size_t ws_size,
                              hipStream_t stream)
{
    (void)in_sizes; (void)n_in; (void)out_size; (void)ws_size;
    const float* emb   = (const float*)d_in[0];
    const float* Wz_w  = (const float*)d_in[1];
    const float* Wz_b  = (const float*)d_in[2];
    const float* Wr_w  = (const float*)d_in[3];
    const float* Wr_b  = (const float*)d_in[4];
    const float* Ur_w  = (const float*)d_in[5];
    const float* Wh_w  = (const float*)d_in[6];
    const float* Wh_b  = (const float*)d_in[7];
    const float* W_w   = (const float*)d_in[8];
    const float* W_b   = (const float*)d_in[9];
    const float* Wo_w  = (const float*)d_in[10];
    const float* Wo_b  = (const float*)d_in[11];
    const float* U_w   = (const float*)d_in[12];
    const float* U_b   = (const float*)d_in[13];
    const float* Ui_w  = (const float*)d_in[14];
    const float* Ui_b  = (const float*)d_in[15];
    const float* Uo_w  = (const float*)d_in[16];
    const float* Uo_b  = (const float*)d_in[17];
    const float* xtree = (const float*)d_in[18];
    const int* word_ids      = (const int*)d_in[19];
    const int* h_nei_idx     = (const int*)d_in[20];
    const int* o_nei_idx     = (const int*)d_in[21];
    const int* contexts      = (const int*)d_in[22];
    const int* pred_targets  = (const int*)d_in[23];
    const int* direction     = (const int*)d_in[24];
    const int* root_word_ids = (const int*)d_in[25];
    const int* root_o_idx    = (const int*)d_in[26];

    float* ws = (float*)d_ws;
    size_t off = 0;
    float* hbuf    = ws + off; off += (size_t)(KE + 1) * KH;      // message buffer
    float* stopbuf = ws + off; off += (size_t)KEB * 2 * KH;       // [x|cur_o] rows + root rows
    float* catA    = ws + off; off += (size_t)KN * 2 * KH;        // [x|sum_h] -> [x|gated]
    float* sumh    = ws + off; off += (size_t)KN * KH;
    float* zbuf    = ws + off; off += (size_t)KN * KH;
    float* rxbuf   = ws + off; off += (size_t)KN * KH;
    float* hnei    = ws + off; off += (size_t)KN * KNB * KH;
    float* rh      = ws + off; off += (size_t)KN * KNB * KH;
    float* h1      = ws + off; off += (size_t)KEB * KH;           // pred hidden / stop s1
    float* bigR    = ws + off; off += (size_t)KEB * KV;           // predcat+scores, later stopcat+s2
    float* accum   = ws + off; off += 16;

    float* predcat = bigR;                      // dead before scores written
    float* scores  = bigR;
    float* stopcat = bigR;                      // reused after pred_loss consumed scores
    float* s2      = bigR + (size_t)KEB * (KH + KL);

    const dim3 blk32(32);
    zero_init<<<1, 256, 0, stream>>>(hbuf, accum);

    for (int t = 0; t < KT; ++t) {
        gru_gather<<<KN, 256, 0, stream>>>(
            emb, hbuf, word_ids + (size_t)t * KN,
            h_nei_idx + (size_t)t * KN * KNB, o_nei_idx + (size_t)t * KN * KNB,
            catA, sumh, hnei, stopbuf + (size_t)t * KN * 2 * KH);
        // z = sigmoid([x|sum_h] @ W_z^T + b)
        gemm64x32<<<dim3(KN / 64, KH / 32), blk32, 0, stream>>>(
            catA, 2 * KH, Wz_w, Wz_b, zbuf, KH, 0, 2 * KH, M_SIG, nullptr, nullptr);
        // rx = x @ W_r^T + b
        gemm64x32<<<dim3(KN / 64, KH / 32), blk32, 0, stream>>>(
            catA, 2 * KH, Wr_w, Wr_b, rxbuf, KH, 0, KH, M_NONE, nullptr, nullptr);
        // rh = sigmoid(rx + h_nei @ U_r^T) * h_nei       (bulk of the FLOPs)
        gemm64x32<<<dim3(KN * KNB / 64, KH / 32), blk32, 0, stream>>>(
            hnei, KH, Ur_w, nullptr, rh, KH, 0, KH, M_RGATE, rxbuf, hnei);
        gated_reduce<<<KN, 256, 0, stream>>>(rh, catA);
        // new_h = (1-z)*sum_h + z*tanh([x|gated] @ W_h^T + b) -> hbuf[t*N ..]
        gemm64x32<<<dim3(KN / 64, KH / 32), blk32, 0, stream>>>(
            catA, 2 * KH, Wh_w, Wh_b, hbuf, KH, t * KN, 2 * KH, M_GRU, zbuf, sumh);
    }

    // ---- word prediction head
    pred_gather<<<KEB, 384, 0, stream>>>(hbuf, xtree, contexts, predcat);
    gemm64x32<<<dim3(KEB / 64, KH / 32), blk32, 0, stream>>>(
        predcat, KH + KL, W_w, W_b, h1, KH, 0, KH + KL, M_RELU, nullptr, nullptr);
    gemm64x32<<<dim3(KEB / 64, KV / 32), blk32, 0, stream>>>(
        h1, KH, Wo_w, Wo_b, scores, KV, 0, KH, M_NONE, nullptr, nullptr);
    pred_loss_kernel<<<KEB, 256, 0, stream>>>(
        scores, pred_targets, direction, root_word_ids, accum);

    // ---- stop head
    root_gather<<<KB, 256, 0, stream>>>(emb, hbuf, root_word_ids, root_o_idx, stopbuf);
    gemm64x32<<<dim3(KEB / 64, KH / 32), blk32, 0, stream>>>(
        stopbuf, 2 * KH, Ui_w, Ui_b, h1, KH, 0, 2 * KH, M_RELU, nullptr, nullptr);
    stop_gather<<<KEB, 384, 0, stream>>>(h1, xtree, contexts, stopcat);
    gemm64x32<<<dim3(KEB / 64, KH / 32), blk32, 0, stream>>>(
        stopcat, KH + KL, U_w, U_b, s2, KH, 0, KH + KL, M_RELU, nullptr, nullptr);
    stop_loss_kernel<<<KEB, 256, 0, stream>>>(s2, Uo_w, Uo_b, direction, accum);

    finalize_kernel<<<1, 1, 0, stream>>>(accum, (float*)d_out);
}